// EdgeMPNN_33621003993339
// MI455X (gfx1250) — compile-verified
//
#include <hip/hip_runtime.h>

// ---------------------------------------------------------------------------
// EdgeMPNN for MI455X (gfx1250): fp32 WMMA (V_WMMA_F32_16X16X4_F32) for all
// linears; fused edge kernel (edge-GEMM + gather + relu + scatter-atomic-add)
// so the [1.6M x 64] intermediate never touches memory. H/ACC (25.6MB each)
// are L2-resident on the 192MB L2, so gathers/atomics run at L2 speed.
// ---------------------------------------------------------------------------

typedef __attribute__((ext_vector_type(2))) float v2f;
typedef __attribute__((ext_vector_type(8))) float v8f;

// D = A(16x4,f32) * B(4x16,f32) + C(16x16,f32), wave32.
__device__ __forceinline__ v8f wmma_f32_16x16x4(v2f a, v2f b, v8f c) {
#if defined(__AMDGCN__) && __has_builtin(__builtin_amdgcn_wmma_f32_16x16x4_f32)
  // 8 args: (neg_a, A, neg_b, B, c_mod, C, reuse_a, reuse_b)
  return __builtin_amdgcn_wmma_f32_16x16x4_f32(false, a, false, b, (short)0, c,
                                               false, false);
#else
  // Host pass / non-gfx1250 placeholder so compilation always succeeds.
  c[0] += a[0] * b[0];
  return c;
#endif
}

// H[n,0:64] = (relu_in ? relu(in[n,:]) : in[n,:]) @ W[K,64] + bias[64]
// One 16-node tile per wave; 4 waves per block. n_nodes assumed % 16 == 0.
__global__ void __launch_bounds__(128)
node_linear_kernel(const float* __restrict__ in, const float* __restrict__ W,
                   const float* __restrict__ bias, float* __restrict__ H,
                   int n_nodes, int K, int relu_in) {
  const int lane = threadIdx.x & 31;
  const int tile = blockIdx.x * 4 + (threadIdx.x >> 5);
  const int n_tiles = n_nodes >> 4;
  if (tile >= n_tiles) return;  // wave-uniform: EXEC stays all-1s for WMMA

  const int m  = lane & 15;          // A: row (M);  B/D: column within tile
  const int kh = (lane >> 4) << 1;   // lanes 0-15 -> K=0,1 ; lanes 16-31 -> K=2,3
  const int arow = (tile << 4) + m;

  v8f acc[4] = {};
  for (int k0 = 0; k0 < K; k0 += 4) {
    v2f a;
    float a0 = in[(size_t)arow * K + k0 + kh + 0];
    float a1 = in[(size_t)arow * K + k0 + kh + 1];
    if (relu_in) { a0 = fmaxf(a0, 0.0f); a1 = fmaxf(a1, 0.0f); }
    a[0] = a0; a[1] = a1;
#pragma unroll
    for (int ct = 0; ct < 4; ++ct) {        // 4 x 16-column output tiles = 64
      const int col = (ct << 4) + m;
      v2f b;
      b[0] = W[(k0 + kh + 0) * 64 + col];
      b[1] = W[(k0 + kh + 1) * 64 + col];
      acc[ct] = wmma_f32_16x16x4(a, b, acc[ct]);
    }
  }
  // D layout: VGPR r holds M = r + (lane>=16 ? 8 : 0), N = lane&15
  const int rbase = (tile << 4) + ((lane >> 4) << 3);
#pragma unroll
  for (int ct = 0; ct < 4; ++ct) {
    const int col = (ct << 4) + m;
    const float bv = bias[col];
#pragma unroll
    for (int r = 0; r < 8; ++r)
      H[(size_t)(rbase + r) * 64 + col] = acc[ct][r] + bv;
  }
}

// Fused: for each edge e: msg = relu(edge_attr[e,:]@We + be + H[src[e],:])
//        ACC[dst[e],:] += msg   (global f32 atomics == segment_sum)
// 16-edge tile per wave iteration; grid-stride so the 16 preloaded B
// fragments (We is 4KB) amortize across tiles. n_edges assumed % 16 == 0.
__global__ void __launch_bounds__(128)
edge_mpnn_kernel(const float* __restrict__ EA, const int* __restrict__ src,
                 const int* __restrict__ dst, const float* __restrict__ We,
                 const float* __restrict__ be, const float* __restrict__ H,
                 float* __restrict__ ACC, int n_edges) {
  const int lane  = threadIdx.x & 31;
  const int gwave = blockIdx.x * 4 + (threadIdx.x >> 5);
  const int n_gw  = gridDim.x * 4;
  const int n_tiles = n_edges >> 4;

  const int m  = lane & 15;
  const int kh = (lane >> 4) << 1;

  // Preload B fragments: We[16,64] -> bfrag[kstep][coltile]
  v2f bfrag[4][4];
  float bev[4];
#pragma unroll
  for (int ks = 0; ks < 4; ++ks)
#pragma unroll
    for (int ct = 0; ct < 4; ++ct) {
      const int col = (ct << 4) + m;
      bfrag[ks][ct][0] = We[(ks * 4 + kh + 0) * 64 + col];
      bfrag[ks][ct][1] = We[(ks * 4 + kh + 1) * 64 + col];
    }
#pragma unroll
  for (int ct = 0; ct < 4; ++ct) bev[ct] = be[(ct << 4) + m];

  for (int tile = gwave; tile < n_tiles; tile += n_gw) {
    const int e0   = tile << 4;
    const int arow = e0 + m;
    v8f acc[4] = {};
#pragma unroll
    for (int ks = 0; ks < 4; ++ks) {
      v2f a;
      a[0] = EA[(size_t)arow * 16 + ks * 4 + kh + 0];
      a[1] = EA[(size_t)arow * 16 + ks * 4 + kh + 1];
#pragma unroll
      for (int ct = 0; ct < 4; ++ct)
        acc[ct] = wmma_f32_16x16x4(a, bfrag[ks][ct], acc[ct]);
    }
    // This lane covers edges e0 + (lane>=16 ? 8:0) + r, r in [0,8)
    const int mb = e0 + ((lane >> 4) << 3);
#pragma unroll
    for (int r = 0; r < 8; ++r) {
      const int e  = mb + r;
      const size_t sb = (size_t)src[e] * 64;
      const size_t db = (size_t)dst[e] * 64;
#pragma unroll
      for (int ct = 0; ct < 4; ++ct) {
        const int col = (ct << 4) + m;
        const float msg = fmaxf(acc[ct][r] + bev[ct] + H[sb + col], 0.0f);
        atomicAdd(&ACC[db + col], msg);  // L2-resident: fast path
      }
    }
  }
}

// out[n] = relu(ACC[n,:]) . Wo + bo
__global__ void __launch_bounds__(256)
out_kernel(const float* __restrict__ ACC, const float* __restrict__ Wo,
           const float* __restrict__ bo, float* __restrict__ out, int n_nodes) {
  const int n = blockIdx.x * blockDim.x + threadIdx.x;
  if (n >= n_nodes) return;
  float s = bo[0];
  const float* row = ACC + (size_t)n * 64;
#pragma unroll
  for (int j = 0; j < 64; ++j) s += fmaxf(row[j], 0.0f) * Wo[j];
  out[n] = s;
}

extern "C" void kernel_launch(void* const* d_in, const int* in_sizes, int n_in,
                              void* d_out, int out_size, void* d_ws,
                              size_t ws_size, hipStream_t stream) {
  const float* x   = (const float*)d_in[0];
  const int*   ei  = (const int*)d_in[1];   // [2,E] int32 per harness convention
  const float* ea  = (const float*)d_in[2];
  const float* W1n = (const float*)d_in[3];  const float* b1n = (const float*)d_in[4];
  const float* W1e = (const float*)d_in[5];  const float* b1e = (const float*)d_in[6];
  const float* W2n = (const float*)d_in[7];  const float* b2n = (const float*)d_in[8];
  const float* W2e = (const float*)d_in[9];  const float* b2e = (const float*)d_in[10];
  const float* W3n = (const float*)d_in[11]; const float* b3n = (const float*)d_in[12];
  const float* W3e = (const float*)d_in[13]; const float* b3e = (const float*)d_in[14];
  const float* Wo  = (const float*)d_in[15]; const float* bo  = (const float*)d_in[16];
  float* out = (float*)d_out;

  const int N = in_sizes[0] / 32;
  const int E = in_sizes[2] / 16;
  const int* src = ei;
  const int* dst = ei + E;

  // Workspace: H [N,64] then ACC [N,64]  (2 * 25.6 MB)
  float* H   = (float*)d_ws;
  float* ACC = H + (size_t)N * 64;
  const size_t accBytes = (size_t)N * 64 * sizeof(float);

  const int node_tiles = N >> 4;
  const dim3 nlb((node_tiles + 3) / 4);
  const int edge_blocks = 4096;              // 16384 waves to hide L2 latency
  const dim3 ob((N + 255) / 256);

  // Layer 1
  node_linear_kernel<<<nlb, 128, 0, stream>>>(x, W1n, b1n, H, N, 32, 0);
  hipMemsetAsync(ACC, 0, accBytes, stream);
  edge_mpnn_kernel<<<edge_blocks, 128, 0, stream>>>(ea, src, dst, W1e, b1e, H, ACC, E);
  // Layer 2 (relu on inter-layer activations)
  node_linear_kernel<<<nlb, 128, 0, stream>>>(ACC, W2n, b2n, H, N, 64, 1);
  hipMemsetAsync(ACC, 0, accBytes, stream);
  edge_mpnn_kernel<<<edge_blocks, 128, 0, stream>>>(ea, src, dst, W2e, b2e, H, ACC, E);
  // Layer 3
  node_linear_kernel<<<nlb, 128, 0, stream>>>(ACC, W3n, b3n, H, N, 64, 1);
  hipMemsetAsync(ACC, 0, accBytes, stream);
  edge_mpnn_kernel<<<edge_blocks, 128, 0, stream>>>(ea, src, dst, W3e, b3e, H, ACC, E);
  // Readout
  out_kernel<<<ob, 256, 0, stream>>>(ACC, Wo, bo, out, N);
}